// SizeWeightedFocalLoss_49306224558105
// MI455X (gfx1250) — compile-verified
//
#include <hip/hip_runtime.h>

// ---------------------------------------------------------------------------
// Size-weighted sigmoid focal loss, reduced to a scalar.
// Bandwidth-bound (128 MiB in / 4 B out -> ~5.6 us at 23.3 TB/s).
// CDNA5 path: V_WMMA_F32_16X16X4_F32 with all-ones B performs the f32
// wave-wide reduction (exact for the integer count path).
// ---------------------------------------------------------------------------

typedef __attribute__((ext_vector_type(2))) float v2f;
typedef __attribute__((ext_vector_type(8))) float v8f;

#define SAMPLES           64
#define ELEMS_PER_SAMPLE  262144   // 512*512
#define BLOCKS_PER_SAMPLE 128
#define CHUNK             2048     // ELEMS_PER_SAMPLE / BLOCKS_PER_SAMPLE
#define THREADS1          256

// Wave32-wide f32 sum via one WMMA (B = ones).
// A layout (32-bit 16x4): VGPR0 lanes 0-15 -> K=0, lanes 16-31 -> K=2.
// With a.y = 0, D[m,n] = p_m + p_{m+16}. Summing the 8 D VGPRs per lane
// leaves rows 0-7 in lanes 0-15 and rows 8-15 in lanes 16-31; shfl_xor(16)
// completes the total (present in every lane).
__device__ __forceinline__ float wave_reduce_sum_wmma(float p) {
  v2f a; a[0] = p;    a[1] = 0.0f;
  v2f b; b[0] = 1.0f; b[1] = 1.0f;
  v8f c = {};
  c = __builtin_amdgcn_wmma_f32_16x16x4_f32(
      /*neg_a=*/false, a, /*neg_b=*/false, b,
      /*c_mod=*/(short)0, c, /*reuse_a=*/false, /*reuse_b=*/false);
  float s = ((c[0] + c[1]) + (c[2] + c[3])) + ((c[4] + c[5]) + (c[6] + c[7]));
  s += __shfl_xor(s, 16, 32);
  return s;
}

// Per-element focal math (one exp, one log, one divide).
__device__ __forceinline__ void focal_elem(float x, int ti,
                                           float& facc, float& cacc) {
  const float t     = (ti > 0) ? 1.0f : 0.0f;
  const float e     = __expf(-fabsf(x));       // exp(-|x|)
  const float denom = 1.0f + e;
  const float inv   = 1.0f / denom;
  const float sig   = (x >= 0.0f) ? inv : e * inv;   // sigmoid(x)
  const float bce   = fmaxf(x, 0.0f) - x * t + __logf(denom);
  const float ompt  = (t == 1.0f) ? (1.0f - sig) : sig;  // 1 - pt
  const float alpha = (t == 1.0f) ? 0.25f : 0.75f;
  facc += alpha * (ompt * ompt) * bce;
  cacc += t;
}

__global__ __launch_bounds__(THREADS1)
void focal_partial_kernel(const float* __restrict__ pred,
                          const int*   __restrict__ tgt,
                          float* __restrict__ partials) {
  const int  blk  = blockIdx.x;
  const int  s    = blockIdx.y;
  const long base = (long)s * ELEMS_PER_SAMPLE + (long)blk * CHUNK;

  const float4* p4 = reinterpret_cast<const float4*>(pred + base);
  const int4*   t4 = reinterpret_cast<const int4*>(tgt + base);

  float facc = 0.0f, cacc = 0.0f;
#pragma unroll
  for (int it = 0; it < 2; ++it) {          // 2 * 256 * 4 = 2048 elems/block
    const int    idx = it * THREADS1 + threadIdx.x;
    const float4 xv  = p4[idx];
    const int4   tv  = t4[idx];
    focal_elem(xv.x, tv.x, facc, cacc);
    focal_elem(xv.y, tv.y, facc, cacc);
    focal_elem(xv.z, tv.z, facc, cacc);
    focal_elem(xv.w, tv.w, facc, cacc);
  }

  // Wave-wide reduction via WMMA (all 256 threads active -> EXEC all ones).
  const float fsum = wave_reduce_sum_wmma(facc);
  const float csum = wave_reduce_sum_wmma(cacc);

  __shared__ float red[16];
  const int wave = threadIdx.x >> 5;
  const int lane = threadIdx.x & 31;
  if (lane == 0) { red[wave] = fsum; red[8 + wave] = csum; }
  __syncthreads();
  if (threadIdx.x == 0) {
    float F = 0.0f, C = 0.0f;
#pragma unroll
    for (int w = 0; w < 8; ++w) { F += red[w]; C += red[8 + w]; }
    const int o = 2 * (s * BLOCKS_PER_SAMPLE + blk);
    partials[o]     = F;
    partials[o + 1] = C;
  }
}

__global__ __launch_bounds__(BLOCKS_PER_SAMPLE)
void per_sample_kernel(const float* __restrict__ partials,
                       float* __restrict__ ps) {
  const int s = blockIdx.x;
  const int t = threadIdx.x;                 // 128 threads = 128 partials
  const int o = 2 * (s * BLOCKS_PER_SAMPLE + t);

  float f = partials[o];
  float c = partials[o + 1];
  f = wave_reduce_sum_wmma(f);
  c = wave_reduce_sum_wmma(c);

  __shared__ float red[8];
  const int wave = t >> 5;
  const int lane = t & 31;
  if (lane == 0) { red[wave] = f; red[4 + wave] = c; }
  __syncthreads();
  if (t == 0) {
    const float F = (red[0] + red[1]) + (red[2] + red[3]);
    const float C = (red[4] + red[5]) + (red[6] + red[7]);
    // fg counts are integers < 2^24: exact in f32 throughout.
    const float sw = (C > 0.0f)
        ? fminf(100.0f / sqrtf(fmaxf(C, 1.0f)), 10.0f)
        : 1.0f;
    ps[s] = (F * (1.0f / (float)ELEMS_PER_SAMPLE)) * sw;
  }
}

__global__ __launch_bounds__(64)
void final_mean_kernel(const float* __restrict__ ps,
                       float* __restrict__ out) {
  float v = ps[threadIdx.x];                 // 64 threads = 2 full waves
  v = wave_reduce_sum_wmma(v);
  __shared__ float r[2];
  if ((threadIdx.x & 31) == 0) r[threadIdx.x >> 5] = v;
  __syncthreads();
  if (threadIdx.x == 0) out[0] = (r[0] + r[1]) * (1.0f / (float)SAMPLES);
}

extern "C" void kernel_launch(void* const* d_in, const int* in_sizes, int n_in,
                              void* d_out, int out_size, void* d_ws, size_t ws_size,
                              hipStream_t stream) {
  (void)in_sizes; (void)n_in; (void)out_size; (void)ws_size;

  const float* pred = (const float*)d_in[0];   // (64,1,512,512) f32
  const int*   tgt  = (const int*)d_in[1];     // (64,512,512) i32
  float*       out  = (float*)d_out;           // scalar f32

  float* partials = (float*)d_ws;                                   // 2*64*128 f32
  float* ps       = partials + 2 * SAMPLES * BLOCKS_PER_SAMPLE;     // 64 f32

  dim3 grid1(BLOCKS_PER_SAMPLE, SAMPLES);
  focal_partial_kernel<<<grid1, THREADS1, 0, stream>>>(pred, tgt, partials);
  per_sample_kernel<<<SAMPLES, BLOCKS_PER_SAMPLE, 0, stream>>>(partials, ps);
  final_mean_kernel<<<1, 64, 0, stream>>>(ps, out);
}